// VelocityBarrier_16664473108528
// MI455X (gfx1250) — compile-verified
//
#include <hip/hip_runtime.h>
#include <hip/hip_bf16.h>
#include <stdint.h>

// ---------------------------------------------------------------------------
// VelocityBarrier: out[b] = sum_d active[d] * ( [xo>=hi] w[d]*(xo-hi)^p
//                                             + [xo<=lo] w[d]*(xo-lo)^p )
// hi = xn + b_lim[d][1], lo = xn + b_lim[d][0].
// Memory-bound streaming kernel: 192 MB in / 16 MB out per call.
// Data path: gfx1250 async DMA global->LDS (ASYNCcnt), double-buffered.
// ---------------------------------------------------------------------------

constexpr int D                 = 6;
constexpr int BLOCK             = 256;
constexpr int ROWS_PER_THREAD   = 2;
constexpr int ROWS_PER_TILE     = BLOCK * ROWS_PER_THREAD;   // 512
constexpr int TILE_F            = ROWS_PER_TILE * D;         // 3072 floats = 12 KB
constexpr int CHUNKS_PER_THREAD = TILE_F / 4 / BLOCK;        // 3 x 16B per array
constexpr int ASYNC_PER_TILE    = 2 * CHUNKS_PER_THREAD;     // 6 async ops / thread / tile

// 16-byte async DMA: global -> LDS, tracked by ASYNCcnt (CDNA5 path).
__device__ __forceinline__ void async_copy_b128(uint32_t lds_addr, const float* gsrc) {
  asm volatile("global_load_async_to_lds_b128 %0, %1, off"
               :: "v"(lds_addr), "v"((uint64_t)(uintptr_t)gsrc)
               : "memory");
}

template <int N>
__device__ __forceinline__ void wait_asynccnt() {
  asm volatile("s_wait_asynccnt %0" :: "n"(N) : "memory");
}

// Integer-exponent pow (matches x**float(p) for integral p on masked domain).
__device__ __forceinline__ float ipow(float b, int e) {
  float r = 1.0f;
  while (e > 0) { if (e & 1) r *= b; b *= b; e >>= 1; }
  return r;
}

__global__ void __launch_bounds__(BLOCK)
velocity_barrier_kernel(const float* __restrict__ x_old,
                        const float* __restrict__ x_new,
                        const float* __restrict__ b_lim,
                        const float* __restrict__ weight,
                        const float* __restrict__ active,
                        const int*   __restrict__ power,
                        float*       __restrict__ out,
                        long long B, int ntiles)
{
  __shared__ __align__(16) float ldsO[2][TILE_F];
  __shared__ __align__(16) float ldsN[2][TILE_F];
  __shared__ __align__(16) float ldsTrash[4];   // sink for clamped tail chunks

  const int tid = threadIdx.x;
  const long long totalF = B * (long long)D;

  // Per-dim parameters: wave-uniform loads, hoisted once per thread.
  float pLo[D], pHi[D], pW[D], pA[D];
  int   pP[D];
#pragma unroll
  for (int d = 0; d < D; ++d) {
    pLo[d] = b_lim[2 * d + 0];
    pHi[d] = b_lim[2 * d + 1];
    pW[d]  = weight[d];
    pA[d]  = active[d];
    pP[d]  = power[d];
  }

  // Stage one tile (both arrays) into LDS buffer `buf` via async DMA.
  // Always issues exactly ASYNC_PER_TILE ops per thread (tail chunks are
  // clamped in-bounds and routed to a trash slot) so the s_wait_asynccnt
  // immediates stay constant.
  auto stage = [&](int t, int buf) {
    const long long tbase = (long long)t * TILE_F;
#pragma unroll
    for (int k = 0; k < CHUNKS_PER_THREAD; ++k) {
      const int c = tid + k * BLOCK;                 // chunk id within tile
      long long goff = tbase + 4ll * c;              // float offset, 16B aligned
      const bool ok  = (goff + 4) <= totalF;         // totalF % 4 == 0
      const long long so = ok ? goff : (totalF - 4);
      const uint32_t dA = (uint32_t)(uintptr_t)(ok ? &ldsO[buf][4 * c] : &ldsTrash[0]);
      const uint32_t dB = (uint32_t)(uintptr_t)(ok ? &ldsN[buf][4 * c] : &ldsTrash[0]);
      async_copy_b128(dA, x_old + so);
      async_copy_b128(dB, x_new + so);
    }
  };

  int t = blockIdx.x;
  const int stride = gridDim.x;
  int cur = 0;
  if (t < ntiles) stage(t, 0);

  for (; t < ntiles; t += stride) {
    const bool pf = (t + stride) < ntiles;
    if (pf) {
      stage(t + stride, cur ^ 1);          // prefetch next tile (in flight)
      wait_asynccnt<ASYNC_PER_TILE>();     // current tile's copies done (in-order)
    } else {
      wait_asynccnt<0>();                  // drain
    }
    __syncthreads();                       // all waves' DMA for `cur` visible

    // --- compute two rows from LDS (3x ds_load_b128 per array) ---
    const float4* o4 = (const float4*)&ldsO[cur][12 * tid];
    const float4* n4 = (const float4*)&ldsN[cur][12 * tid];
    const float4 oa = o4[0], ob = o4[1], oc = o4[2];
    const float4 na = n4[0], nb = n4[1], nc = n4[2];
    const float xo[12] = {oa.x, oa.y, oa.z, oa.w, ob.x, ob.y,
                          ob.z, ob.w, oc.x, oc.y, oc.z, oc.w};
    const float xn[12] = {na.x, na.y, na.z, na.w, nb.x, nb.y,
                          nb.z, nb.w, nc.x, nc.y, nc.z, nc.w};

    float acc0 = 0.0f, acc1 = 0.0f;
#pragma unroll
    for (int d = 0; d < D; ++d) {
      {
        const float hi = xn[d] + pHi[d], lo = xn[d] + pLo[d];
        const float dh = xo[d] - hi,     dl = xo[d] - lo;
        const float up = (xo[d] >= hi) ? pW[d] * ipow(dh, pP[d]) : 0.0f;
        const float dn = (xo[d] <= lo) ? pW[d] * ipow(dl, pP[d]) : 0.0f;
        acc0 += pA[d] * (up + dn);
      }
      {
        const float hi = xn[D + d] + pHi[d], lo = xn[D + d] + pLo[d];
        const float dh = xo[D + d] - hi,     dl = xo[D + d] - lo;
        const float up = (xo[D + d] >= hi) ? pW[d] * ipow(dh, pP[d]) : 0.0f;
        const float dn = (xo[D + d] <= lo) ? pW[d] * ipow(dl, pP[d]) : 0.0f;
        acc1 += pA[d] * (up + dn);
      }
    }

    const long long r0 = (long long)t * ROWS_PER_TILE + 2 * tid;
    if (r0 + 1 < B) {
      float2 v; v.x = acc0; v.y = acc1;
      *(float2*)(out + r0) = v;            // 8B-aligned coalesced store
    } else if (r0 < B) {
      out[r0] = acc0;
    }

    __syncthreads();                       // protect `cur^1` before restage
    cur ^= 1;
  }
}

extern "C" void kernel_launch(void* const* d_in, const int* in_sizes, int n_in,
                              void* d_out, int out_size, void* d_ws, size_t ws_size,
                              hipStream_t stream) {
  const float* x_old  = (const float*)d_in[0];
  const float* x_new  = (const float*)d_in[1];
  const float* b_lim  = (const float*)d_in[2];
  const float* weight = (const float*)d_in[3];
  const float* active = (const float*)d_in[4];
  const int*   power  = (const int*)d_in[5];
  float*       out    = (float*)d_out;

  const long long totalF = (long long)in_sizes[0];
  const long long B      = totalF / D;
  int ntiles = (int)((B + ROWS_PER_TILE - 1) / ROWS_PER_TILE);
  if (ntiles < 1) ntiles = 1;
  int blocks = ntiles < 4096 ? ntiles : 4096;

  velocity_barrier_kernel<<<blocks, BLOCK, 0, stream>>>(
      x_old, x_new, b_lim, weight, active, power, out, B, ntiles);
}